// TwoPointSpatialStatsLoss_62689342652632
// MI455X (gfx1250) — compile-verified
//
#include <hip/hip_runtime.h>
#include <math.h>

// ---------------------------------------------------------------------------
// TwoPointSpatialStatsLoss for MI455X (gfx1250).
// 2D FFT autocorrelation pipeline; each 512-point FFT is a four-step
// 16x32 factorization whose DFT-16 / DFT-32 stages run on V_WMMA_F32_16X16X4_F32.
// Memory-bound (~5 GB traffic -> ~220us floor at 23.3 TB/s); passes are fused
// (exp into pass1, |F|^2 into pass2, roll+minmax into pass4, transposes into
// the FFT stores via LDS) to stay near that floor.
// Twiddles use hardware v_sin/v_cos (phases are pre-reduced to [-2pi, 2pi]).
// ---------------------------------------------------------------------------

#define EL 512
#define ROWS_PER_WG 8
#define THREADS 256
#define REGION 2120              // per-wave LDS floats; 2120 % 64 == 8 -> conflict-free coop store
#define XIN_OFF 0                // 512 float2 (staging, later X output)
#define ARE_OFF 1032             // 32x16 A'-real
#define AIM_OFF 1556             // 32x16 A'-imag

typedef float v2f __attribute__((ext_vector_type(2)));
typedef float v8f __attribute__((ext_vector_type(8)));

// D = A(16x4) * B(4x16) + C, f32. (neg_a/neg_b must be false for F32 per ISA NEG table)
__device__ __forceinline__ v8f wmma4(v2f a, v2f b, v8f c) {
  return __builtin_amdgcn_wmma_f32_16x16x4_f32(false, a, false, b, (short)0, c, false, false);
}

// MODE 1: real in (exp of d_in rows), fwd FFT, complex transposed out
// MODE 2: complex in, fwd FFT, |.|^2 * scale real transposed out
// MODE 3: real in (power), inv FFT, complex transposed out
// MODE 4: complex in, inv FFT, real part transposed+rolled out, per-block min/max
template <int MODE>
__global__ __launch_bounds__(THREADS)
void fft512_pass(const float* __restrict__ rinA, const float* __restrict__ rinB,
                 const float2* __restrict__ cin, float2* __restrict__ cout,
                 float* __restrict__ rout, float* __restrict__ mmPart)
{
  __shared__ float lds[ROWS_PER_WG * REGION];
  __shared__ float red[2 * THREADS];

  const int tid  = threadIdx.x;
  const int wave = tid >> 5;
  const int lane = tid & 31;
  const int lo = lane & 15, hi = lane >> 4;
  float* W = &lds[wave * REGION];

  const int gRow0 = blockIdx.x * ROWS_PER_WG;   // 8 rows per WG, never straddles an image
  const int myRow = gRow0 + wave;

  constexpr bool INV = (MODE >= 3);
  constexpr bool CPLX_IN = (MODE == 2 || MODE == 4);
  const float s = INV ? 1.0f : -1.0f;
  const float TWO_PI = 6.283185307179586f;

  // --- twiddle operands in registers (A-operand layout: m=lane&15, k=4c+2hi+j) ---
  // phases are integer-reduced mod N, so |ang| < 2pi: hardware v_sin/v_cos is exact enough
  float w16re[4][2], w16im[4][2];
#pragma unroll
  for (int c = 0; c < 4; ++c)
#pragma unroll
    for (int j = 0; j < 2; ++j) {
      int kk = 4 * c + 2 * hi + j;
      int ph = (lo * kk) & 15;
      float ang = s * TWO_PI * (float)ph * (1.0f / 16.0f);
      w16re[c][j] = __cosf(ang); w16im[c][j] = __sinf(ang);
    }
  float w32re[2][8][2], w32im[2][8][2];
#pragma unroll
  for (int mt = 0; mt < 2; ++mt)
#pragma unroll
    for (int c = 0; c < 8; ++c)
#pragma unroll
      for (int j = 0; j < 2; ++j) {
        int k2 = mt * 16 + lo;
        int n2 = 4 * c + 2 * hi + j;
        int ph = (k2 * n2) & 31;
        float ang = s * TWO_PI * (float)ph * (1.0f / 32.0f);
        w32re[mt][c][j] = __cosf(ang); w32im[mt][c][j] = __sinf(ang);
      }

  // --- stage one row (coalesced) into this wave's LDS region ---
  if (!CPLX_IN) {
    const float* src; long base;
    if (MODE == 1) { src = (myRow < 32768) ? rinA : rinB; base = (long)(myRow & 32767) * EL; }
    else           { src = rinA;                          base = (long)myRow * EL; }
#pragma unroll
    for (int i = 0; i < 16; ++i) {
      float v = src[base + lane + 32 * i];
      if (MODE == 1) v = __expf(-8.0f * v * v);   // soft_equality, 2*eps^2 = 0.125
      W[XIN_OFF + lane + 32 * i] = v;
    }
  } else {
    long base = (long)myRow * EL;
#pragma unroll
    for (int i = 0; i < 16; ++i) {
      float2 v = cin[base + lane + 32 * i];
      W[XIN_OFF + 2 * (lane + 32 * i) + 0] = v.x;
      W[XIN_OFF + 2 * (lane + 32 * i) + 1] = v.y;
    }
  }
  __syncthreads();

  // --- Step 1: A[k1][n2] = sum_{n1<16} W16[k1][n1] * x[32*n1+n2]  (WMMA, N=32 in 2 tiles)
  //     Step 2: A'[k1][n2] = A * e^{s*2pi*i*n2*k1/512}, stored transposed to LDS ---
#pragma unroll
  for (int t = 0; t < 2; ++t) {
    v8f dre = {}, dim = {};
#pragma unroll
    for (int c = 0; c < 4; ++c) {
      v2f are, aim, bre, bim, bimn;
      are[0] = w16re[c][0]; are[1] = w16re[c][1];
      aim[0] = w16im[c][0]; aim[1] = w16im[c][1];
#pragma unroll
      for (int j = 0; j < 2; ++j) {
        int n1 = 4 * c + 2 * hi + j;
        int nidx = 32 * n1 + t * 16 + lo;
        if (CPLX_IN) {
          bre[j] = W[XIN_OFF + 2 * nidx + 0];
          bim[j] = W[XIN_OFF + 2 * nidx + 1];
        } else {
          bre[j] = W[XIN_OFF + nidx];
        }
      }
      dre = wmma4(are, bre, dre);
      dim = wmma4(aim, bre, dim);
      if (CPLX_IN) {
        bimn[0] = -bim[0]; bimn[1] = -bim[1];
        dre = wmma4(aim, bimn, dre);   // -Wim*Bim
        dim = wmma4(are, bim,  dim);
      }
    }
#pragma unroll
    for (int r = 0; r < 8; ++r) {          // D layout: M = r + 8*hi, N = lo
      int k1 = r + 8 * hi;
      int n2 = t * 16 + lo;
      int ph = (n2 * k1) & 511;
      float ang = s * TWO_PI * (float)ph * (1.0f / 512.0f);
      float cc = __cosf(ang), ss = __sinf(ang);
      float re = dre[r], im = dim[r];
      W[ARE_OFF + n2 * 16 + k1] = re * cc - im * ss;
      W[AIM_OFF + n2 * 16 + k1] = re * ss + im * cc;
    }
  }
  __syncthreads();

  // --- Step 3: X[k1+16*k2] = sum_{n2<32} W32[k2][n2] * A'[n2][k1]  (WMMA, M=32 in 2 tiles)
#pragma unroll
  for (int mt = 0; mt < 2; ++mt) {
    v8f ere = {}, eim = {};
#pragma unroll
    for (int c = 0; c < 8; ++c) {
      v2f are, aim, bre, bim, bimn;
      are[0] = w32re[mt][c][0]; are[1] = w32re[mt][c][1];
      aim[0] = w32im[mt][c][0]; aim[1] = w32im[mt][c][1];
#pragma unroll
      for (int j = 0; j < 2; ++j) {
        int n2 = 4 * c + 2 * hi + j;
        bre[j] = W[ARE_OFF + n2 * 16 + lo];
        bim[j] = W[AIM_OFF + n2 * 16 + lo];
      }
      bimn[0] = -bim[0]; bimn[1] = -bim[1];
      ere = wmma4(are, bre,  ere);
      ere = wmma4(aim, bimn, ere);
      eim = wmma4(are, bim,  eim);
      eim = wmma4(aim, bre,  eim);
    }
#pragma unroll
    for (int r = 0; r < 8; ++r) {
      int k2 = mt * 16 + r + 8 * hi;
      int k  = lo + 16 * k2;
      W[XIN_OFF + 2 * k + 0] = ere[r];
      W[XIN_OFF + 2 * k + 1] = eim[r];
    }
  }
  __syncthreads();

  // --- cooperative transposed store: out[k][row], 64B contiguous segments ---
  const int imgW = gRow0 >> 9;
  const int r0   = gRow0 & 511;
  const int rj   = tid & 7;
  const int kk0  = tid >> 3;
  const float* Wr = &lds[rj * REGION];
  float myMn = 1e30f, myMx = -1e30f;
#pragma unroll
  for (int it = 0; it < 16; ++it) {
    int k = kk0 + 32 * it;
    float re = Wr[XIN_OFF + 2 * k + 0];
    float im = Wr[XIN_OFF + 2 * k + 1];
    long obase = (long)imgW * (long)(EL * EL);
    if (MODE == 1 || MODE == 3) {
      float2 v; v.x = re; v.y = im;
      cout[obase + (long)k * EL + r0 + rj] = v;
    } else if (MODE == 2) {
      // |F|^2/el^2 (power) * 1/el^2 (ifft2 norm) = |F|^2 / 512^4
      rout[obase + (long)k * EL + r0 + rj] = (re * re + im * im) * 1.4551915228366852e-11f;
    } else { // MODE 4: real ac with roll(256,256): (i+256)%512 == i^256
      rout[obase + (long)(k ^ 256) * EL + ((r0 + rj) ^ 256)] = re;
      myMn = fminf(myMn, re); myMx = fmaxf(myMx, re);
    }
  }
  if (MODE == 4) {
    red[tid] = myMn; red[THREADS + tid] = myMx;
    __syncthreads();
    for (int off = THREADS / 2; off > 0; off >>= 1) {
      if (tid < off) {
        red[tid] = fminf(red[tid], red[tid + off]);
        red[THREADS + tid] = fmaxf(red[THREADS + tid], red[THREADS + tid + off]);
      }
      __syncthreads();
    }
    if (tid == 0) {
      int blkInImg = (gRow0 & 511) >> 3;               // 0..63
      mmPart[(imgW * 64 + blkInImg) * 2 + 0] = red[0];
      mmPart[(imgW * 64 + blkInImg) * 2 + 1] = red[THREADS];
    }
  }
}

__global__ void minmax_finalize(const float* __restrict__ mmPart, float* __restrict__ mm) {
  __shared__ float red[128];
  int img = blockIdx.x, t = threadIdx.x;   // 64 threads
  red[t]      = mmPart[(img * 64 + t) * 2 + 0];
  red[64 + t] = mmPart[(img * 64 + t) * 2 + 1];
  __syncthreads();
  for (int off = 32; off > 0; off >>= 1) {
    if (t < off) {
      red[t]      = fminf(red[t], red[t + off]);
      red[64 + t] = fmaxf(red[64 + t], red[64 + t + off]);
    }
    __syncthreads();
  }
  if (t == 0) { mm[img * 2 + 0] = red[0]; mm[img * 2 + 1] = red[64]; }
}

__global__ __launch_bounds__(256)
void finalize_k(const float* __restrict__ ac, const float* __restrict__ mm,
                const float* __restrict__ inA, const float* __restrict__ inB,
                float* __restrict__ out, float* __restrict__ diffPart) {
  __shared__ float red[256];
  int blk = blockIdx.x;             // 64*512
  int b = blk >> 9, y = blk & 511;
  int t = threadIdx.x;
  float mnI = mm[b * 2], sI = 1.0f / (mm[b * 2 + 1] - mnI + 1e-12f);
  float mnT = mm[(64 + b) * 2], sT = 1.0f / (mm[(64 + b) * 2 + 1] - mnT + 1e-12f);
  float dy = (float)(y - 256);
  long rowIn = ((long)b * EL + y) * EL;
  long rowTg = ((long)(64 + b) * EL + y) * EL;
  long oIn = 1 + ((long)b * EL + y) * 1024;
  long oTg = oIn + (long)64 * EL * 1024;
  float acc = 0.0f;
#pragma unroll
  for (int i = 0; i < 2; ++i) {
    int x = t + 256 * i;
    float dx = (float)(x - 256);
    float msk = __expf(-(dx * dx + dy * dy) * (1.0f / 800.0f));   // 2*rad^2 = 800
    float ia = (ac[rowIn + x] - mnI) * sI * msk;
    float ta = (ac[rowTg + x] - mnT) * sT * msk;
    out[oIn + x]       = inA[rowIn + x];
    out[oIn + 512 + x] = ia;
    out[oTg + x]       = inB[rowIn + x];
    out[oTg + 512 + x] = ta;
    float d = ia - ta; acc += d * d;
  }
  red[t] = acc; __syncthreads();
  for (int off = 128; off > 0; off >>= 1) { if (t < off) red[t] += red[t + off]; __syncthreads(); }
  if (t == 0) diffPart[blk] = red[0];
}

__global__ void diff_finalize(const float* __restrict__ diffPart, float* __restrict__ out) {
  __shared__ float red[256];
  int t = threadIdx.x;
  float s = 0.0f;
  for (int i = t; i < 32768; i += 256) s += diffPart[i];
  red[t] = s; __syncthreads();
  for (int off = 128; off > 0; off >>= 1) { if (t < off) red[t] += red[t + off]; __syncthreads(); }
  if (t == 0) out[0] = red[0] * (1.0f / 16777216.0f);   // mean over 64*512*512
}

extern "C" void kernel_launch(void* const* d_in, const int* in_sizes, int n_in,
                              void* d_out, int out_size, void* d_ws, size_t ws_size,
                              hipStream_t stream) {
  const float* inA = (const float*)d_in[0];
  const float* inB = (const float*)d_in[1];
  float* out = (float*)d_out;
  char* ws = (char*)d_ws;
  // workspace map (needs ~385 MB): 256MB complex ping + 128MB real (power, then ac) + scratch
  float2* cbuf    = (float2*)(ws);
  float*  rbuf    = (float*)(ws + 268435456ull);
  float*  mmPart  = (float*)(ws + 268435456ull + 134217728ull);
  float*  mmFinal = (float*)(ws + 268435456ull + 134217728ull + 65536ull);
  float*  dpart   = (float*)(ws + 268435456ull + 134217728ull + 65536ull + 1024ull);

  const int G = (128 * EL) / ROWS_PER_WG;   // 8192 WGs per FFT pass
  fft512_pass<1><<<G, THREADS, 0, stream>>>(inA, inB, nullptr, cbuf, nullptr, nullptr);
  fft512_pass<2><<<G, THREADS, 0, stream>>>(nullptr, nullptr, cbuf, nullptr, rbuf, nullptr);
  fft512_pass<3><<<G, THREADS, 0, stream>>>(rbuf, nullptr, nullptr, cbuf, nullptr, nullptr);
  fft512_pass<4><<<G, THREADS, 0, stream>>>(nullptr, nullptr, cbuf, nullptr, rbuf, mmPart);
  minmax_finalize<<<128, 64, 0, stream>>>(mmPart, mmFinal);
  finalize_k<<<64 * EL, 256, 0, stream>>>(rbuf, mmFinal, inA, inB, out, dpart);
  diff_finalize<<<1, 256, 0, stream>>>(dpart, out);
}